// GraphConvolutionLayer_20633022890739
// MI455X (gfx1250) — compile-verified
//
#include <hip/hip_runtime.h>

// ---------------------------------------------------------------------------
// diag[b,t,n] = sum_d x[b,t,n,d]*adj[b,t,n,d];  out[...,k] = diag*W[k] + b[k]
// Memory-bound (944 MB @ 23.3 TB/s ~ 40us). Streaming NT float4 loads for the
// reduction; V_WMMA_F32_16X16X4_F32 for the rank-1 fc epilogue.
// ---------------------------------------------------------------------------

typedef __attribute__((ext_vector_type(2))) float v2f;
typedef __attribute__((ext_vector_type(4))) float f32x4;
typedef __attribute__((ext_vector_type(8))) float v8f;

#define BT_DIM        (8 * 24)
#define N_DIM         768
#define K_OUT         64
#define TOTAL_ROWS    (BT_DIM * N_DIM)        /* 147456 */
#define ROWS_PER_WAVE 16
#define WAVES_PER_BLK 8
#define THREADS_PER_BLK (WAVES_PER_BLK * 32)  /* 256 = 8 wave32 */
#define NUM_BLOCKS    (TOTAL_ROWS / (ROWS_PER_WAVE * WAVES_PER_BLK)) /* 1152 */

__global__ __launch_bounds__(THREADS_PER_BLK)
void gcn_diag_fc_kernel(const float* __restrict__ x,
                        const float* __restrict__ adj,
                        const float* __restrict__ W,     // [64,1] flat
                        const float* __restrict__ bias,  // [64]
                        float* __restrict__ out)         // [rows,64]
{
    const int lane    = threadIdx.x & 31;
    const int wave    = threadIdx.x >> 5;
    const int rowBase = (blockIdx.x * WAVES_PER_BLK + wave) * ROWS_PER_WAVE;

    // Lane m (m < 16) accumulates diag[rowBase + m]; upper lanes stay 0.
    float diagLane = 0.0f;

    for (int r = 0; r < ROWS_PER_WAVE; ++r) {
        const int row = rowBase + r;
        const f32x4* __restrict__ xr =
            (const f32x4*)(x   + (size_t)row * N_DIM);
        const f32x4* __restrict__ ar =
            (const f32x4*)(adj + (size_t)row * N_DIM);

        // Prefetch next row (24 x 128B lines per tensor; 96B lane stride).
        if (r + 1 < ROWS_PER_WAVE) {
            __builtin_prefetch(x   + (size_t)(row + 1) * N_DIM + lane * 24, 0, 1);
            __builtin_prefetch(adj + (size_t)(row + 1) * N_DIM + lane * 24, 0, 1);
        }

        // 768 floats = 6 iters x (32 lanes x float4): fully coalesced B128,
        // non-temporal (read-once stream, don't thrash L2).
        f32x4 acc = {0.0f, 0.0f, 0.0f, 0.0f};
#pragma unroll
        for (int i = 0; i < N_DIM / 128; ++i) {
            f32x4 xv = __builtin_nontemporal_load(&xr[i * 32 + lane]);
            f32x4 av = __builtin_nontemporal_load(&ar[i * 32 + lane]);
            acc += xv * av;   // contracts to v_fma under -O3
        }
        float s = acc.x + acc.y + acc.z + acc.w;

        // Wave32 butterfly: every lane ends with the full row sum.
#pragma unroll
        for (int off = 16; off > 0; off >>= 1)
            s += __shfl_xor(s, off, 32);

        diagLane = (lane == r) ? s : diagLane;   // place diag[r] in lane r
    }

    // -----------------------------------------------------------------------
    // Epilogue: out[m,k] = diag[m]*W[k] + bias[k] for this 16-row block,
    // as D(16x16) = A(16x4) x B(4x16) + C using V_WMMA_F32_16X16X4_F32.
    //   A layout (ISA 7.12.2): v0 lanes0-15 = K=0 col (diag), lanes16-31 = K=2
    //   (zero); v1 = K=1/K=3 (zero).  B: only K=0 row nonzero (= W slice).
    //   C: bias broadcast down each column.
    // EXEC is all-ones here (no divergent branches) as WMMA requires.
    // -----------------------------------------------------------------------
    v2f aop;
    aop.x = (lane < 16) ? diagLane : 0.0f;
    aop.y = 0.0f;

    const int nCol16 = lane & 15;
    const int mOff   = (lane >> 4) * 8;

#pragma unroll
    for (int kg = 0; kg < K_OUT / 16; ++kg) {
        v2f bop;
        bop.x = (lane < 16) ? W[kg * 16 + lane] : 0.0f;  // B[K=0, n] = W[n]
        bop.y = 0.0f;

        const float bv = bias[kg * 16 + nCol16];
        v8f c;
#pragma unroll
        for (int j = 0; j < 8; ++j) c[j] = bv;

        // 8 args: (neg_a, A, neg_b, B, c_mod, C, reuse_a, reuse_b)
        v8f d = __builtin_amdgcn_wmma_f32_16x16x4_f32(
            false, aop, false, bop, (short)0, c, false, false);

        // D layout: VGPR j, lanes0-15 -> M=j, lanes16-31 -> M=j+8, N=lane&15.
        // 16 lanes x 4B = 64B coalesced segments per store.
#pragma unroll
        for (int j = 0; j < 8; ++j) {
            const size_t orow = (size_t)(rowBase + mOff + j);
            __builtin_nontemporal_store(d[j], &out[orow * K_OUT + kg * 16 + nCol16]);
        }
    }
}

extern "C" void kernel_launch(void* const* d_in, const int* in_sizes, int n_in,
                              void* d_out, int out_size, void* d_ws, size_t ws_size,
                              hipStream_t stream) {
    const float* x    = (const float*)d_in[0];
    const float* adj  = (const float*)d_in[1];
    const float* W    = (const float*)d_in[2];  // [64,1]
    const float* bias = (const float*)d_in[3];  // [64]
    float* out        = (float*)d_out;          // [8*24*768, 64]

    gcn_diag_fc_kernel<<<NUM_BLOCKS, THREADS_PER_BLK, 0, stream>>>(
        x, adj, W, bias, out);
}